// JacobianDeterminantLoss_71519795413077
// MI455X (gfx1250) — compile-verified
//
#include <hip/hip_runtime.h>

// ---------------------------------------------------------------------------
// JacobianDeterminantLoss for u: (B=2, C=3, D=160, H=192, W=160) fp32
// out = mean(relu(-det(J)))   (single fp32 scalar)
//
// MI455X reasoning: 118 MB unique traffic @ 23.3 TB/s => ~5us floor; compute
// is ~0.4 GFLOP (irrelevant). Whole tensor fits in 192MB L2, so stencil
// neighbor re-reads are L2 hits. One streaming pass + deterministic two-stage
// reduction. Final stage uses V_WMMA_F32_16X16X4_F32 (fp32-exact MACs) to
// collapse 64 partials per instruction on the matrix pipe.
// ---------------------------------------------------------------------------

#define DD 160
#define HH 192
#define WW 160
#define SX (HH * WW)          // stride along D axis (x)  = 30720
#define SY (WW)               // stride along H axis (y)  = 160
#define SC (DD * HH * WW)     // channel stride           = 4915200
#define SB (3 * SC)           // batch stride
#define NVOX (2 * DD * HH * WW)   // 9,830,400 voxels
#define VOX_PER_THREAD 8
#define BLOCK 256
#define GRID (NVOX / (BLOCK * VOX_PER_THREAD))   // = 4800, no tail, mult of 64

typedef __attribute__((ext_vector_type(2))) float v2f;
typedef __attribute__((ext_vector_type(8))) float v8f;

// jnp.gradient, edge_order=1: interior (a[i+1]-a[i-1])/2, edges one-sided.
__device__ __forceinline__ float gradf(const float* __restrict__ p,
                                       int c, int n, int stride) {
    if (c == 0)       return p[stride] - p[0];
    if (c == n - 1)   return p[0] - p[-stride];
    return (p[stride] - p[-stride]) * 0.5f;
}

__global__ __launch_bounds__(BLOCK)
void jacdet_partial_kernel(const float* __restrict__ u,
                           float* __restrict__ partials) {
    const float sx = 79.5f;   // (D-1)/2
    const float sy = 95.5f;   // (H-1)/2
    const float sz = 79.5f;   // (W-1)/2

    float acc = 0.0f;
    int base = blockIdx.x * (BLOCK * VOX_PER_THREAD) + threadIdx.x;

#pragma unroll
    for (int i = 0; i < VOX_PER_THREAD; ++i) {
        int n = base + i * BLOCK;            // flat voxel id, coalesced per wave
        int z = n % WW;
        int t = n / WW;
        int y = t % HH;
        t /= HH;
        int x = t % DD;
        int b = t / DD;

        const float* p0 = u + (size_t)b * SB + (size_t)x * SX + y * SY + z;
        const float* p1 = p0 + SC;
        const float* p2 = p1 + SC;

        // channel 0 (scale 79.5 folded in: gradient is linear)
        float dxx = fmaf(sx, gradf(p0, x, DD, SX), 1.0f);
        float dxy = sx * gradf(p0, y, HH, SY);
        float dxz = sx * gradf(p0, z, WW, 1);
        // channel 1 (scale 95.5)
        float dyx = sy * gradf(p1, x, DD, SX);
        float dyy = fmaf(sy, gradf(p1, y, HH, SY), 1.0f);
        float dyz = sy * gradf(p1, z, WW, 1);
        // channel 2 (scale 79.5)
        float dzx = sz * gradf(p2, x, DD, SX);
        float dzy = sz * gradf(p2, y, HH, SY);
        float dzz = fmaf(sz, gradf(p2, z, WW, 1), 1.0f);

        float det = dxx * (dyy * dzz - dyz * dzy)
                  + dxy * (dyz * dzx - dyx * dzz)
                  + dxz * (dyx * dzy - dyy * dzx);
        acc += fmaxf(-det, 0.0f);
    }

    // block reduction (deterministic), 256 threads = 8 wave32
    __shared__ float red[BLOCK];
    red[threadIdx.x] = acc;
    __syncthreads();
#pragma unroll
    for (int s = BLOCK / 2; s > 0; s >>= 1) {
        if (threadIdx.x < s) red[threadIdx.x] += red[threadIdx.x + s];
        __syncthreads();
    }
    if (threadIdx.x == 0) partials[blockIdx.x] = red[0];
}

// ---------------------------------------------------------------------------
// Final reduction: one wave32, C += A_chunk(16x4) * ones(4x16).
// A layout (ISA 7.12.2): lane l, vgpr j -> element (m=l%16, k=2*(l/16)+j);
// every (lane,vgpr) slot is a distinct A entry, so loading 2 consecutive
// floats per lane covers a 64-float chunk exactly once. With B = ones,
// D[m][n] = C[m][n] + sum_k A[m][k]; after all chunks each column of C holds
// the 16 row-sums. Column 0 lives in lane 0 (M=0..7) and lane 16 (M=8..15).
// fp32 MACs -> exact fp32 summation order, deterministic.
// ---------------------------------------------------------------------------
__global__ __launch_bounds__(32)
void jacdet_final_wmma_kernel(const float* __restrict__ partials,
                              float* __restrict__ out,
                              int nchunks, float invN) {
    int lane = threadIdx.x;            // 32 lanes, EXEC all-ones (WMMA req.)
    v8f c = {0.f, 0.f, 0.f, 0.f, 0.f, 0.f, 0.f, 0.f};
    v2f ones; ones[0] = 1.0f; ones[1] = 1.0f;

    for (int i = 0; i < nchunks; ++i) {
        v2f a = *(const v2f*)(partials + (size_t)i * 64 + lane * 2);
        // 8 args: (neg_a, A, neg_b, B, c_mod, C, reuse_a, reuse_b)
        c = __builtin_amdgcn_wmma_f32_16x16x4_f32(
                false, a, false, ones, (short)0, c, false, false);
    }

    float t = c[0] + c[1] + c[2] + c[3] + c[4] + c[5] + c[6] + c[7];
    __shared__ float s[32];
    s[lane] = t;
    __syncthreads();
    if (lane == 0) out[0] = (s[0] + s[16]) * invN;
}

extern "C" void kernel_launch(void* const* d_in, const int* in_sizes, int n_in,
                              void* d_out, int out_size, void* d_ws, size_t ws_size,
                              hipStream_t stream) {
    const float* u = (const float*)d_in[0];
    float* out = (float*)d_out;
    float* partials = (float*)d_ws;    // 4800 floats = 19.2 KB scratch

    jacdet_partial_kernel<<<GRID, BLOCK, 0, stream>>>(u, partials);

    const int nchunks = GRID / 64;     // 75 chunks of 64 partials
    jacdet_final_wmma_kernel<<<1, 32, 0, stream>>>(
        partials, out, nchunks, 1.0f / (float)NVOX);
}